// SPDGeometricMean_78915729097348
// MI455X (gfx1250) — compile-verified
//
#include <hip/hip_runtime.h>
#include <math.h>

#define DN 128
#define DD (DN * DN)
#define BB 32
#define NN 64
#define NBM 2048  // N*B matrices

typedef __attribute__((ext_vector_type(2))) float v2f;
typedef __attribute__((ext_vector_type(8))) float v8f;

// ---------------------------------------------------------------------------
// Batched 128x128x128 fp32 GEMM on the CDNA5 WMMA pipe.
//   C[m] = alpha * A[ia] @ B[ib] + diag * I,  ia = modA ? m%modA : m, etc.
// One workgroup (8 wave32) per matrix. K tiled by 32 through LDS.
// A staged row-major (pad 36) -> A frag = one aligned ds_load_b64.
// B staged pair-interleaved BsP[kpair][col][2] -> B frag = one ds_load_b64.
// ---------------------------------------------------------------------------
__global__ __launch_bounds__(256) void k_bgemm(const float* __restrict__ A,
                                               const float* __restrict__ B,
                                               float* __restrict__ C,
                                               int modA, int modB,
                                               float alpha, float diag) {
  __shared__ float As[DN][36];       // pad 36: stride-36 frag reads hit 16 banks
  __shared__ float BsP[16 * 256];    // [kpair 0..15][col 0..127][2]

  const int m  = blockIdx.x;
  const int ia = modA ? (m % modA) : m;
  const int ib = modB ? (m % modB) : m;
  const float* __restrict__ Ag = A + (size_t)ia * DD;
  const float* __restrict__ Bg = B + (size_t)ib * DD;
  float* __restrict__ Cg = C + (size_t)m * DD;

  const int t    = threadIdx.x;
  const int lane = t & 31;
  const int wave = t >> 5;      // tile row 0..7 (16 rows each)
  const int half = lane >> 4;   // lane group selects K pair / C row offset
  const int lm   = lane & 15;
  const int kb   = half << 1;

  v8f acc[8];
  const v8f vzero = {0.f, 0.f, 0.f, 0.f, 0.f, 0.f, 0.f, 0.f};
#pragma unroll
  for (int tc = 0; tc < 8; ++tc) acc[tc] = vzero;

  for (int k0 = 0; k0 < DN; k0 += 32) {
    __syncthreads();
    // Stage A slab: 128 rows x 32 k-cols (1024 float4, 4 per thread)
    for (int i = t; i < 1024; i += 256) {
      int r = i >> 3, c4 = (i & 7) << 2;
      float4 v = *(const float4*)(Ag + (size_t)r * DN + k0 + c4);
      *(float4*)(&As[r][c4]) = v;
    }
    // Stage B slab pair-interleaved: 16 k-pairs x 128 cols.
    // Each item: two row-adjacent float4 loads -> 4 interleaved float2 stores.
    for (int i = t; i < 512; i += 256) {
      int rp = i >> 5;              // k pair 0..15
      int c4 = (i & 31) << 2;       // col group
      const float* src = Bg + (size_t)(k0 + rp * 2) * DN + c4;
      float4 va = *(const float4*)(src);
      float4 vb = *(const float4*)(src + DN);
      float a0[4] = {va.x, va.y, va.z, va.w};
      float b0[4] = {vb.x, vb.y, vb.z, vb.w};
#pragma unroll
      for (int j = 0; j < 4; ++j) {
        float2 p;
        p.x = a0[j];                // k = 2*rp
        p.y = b0[j];                // k = 2*rp + 1
        *(float2*)(&BsP[rp * 256 + (c4 + j) * 2]) = p;
      }
    }
    __syncthreads();

#pragma unroll
    for (int ks = 0; ks < 8; ++ks) {         // 8 x (K=4) per slab
      const int kk = (ks << 2) + kb;         // even -> 8B aligned
      v2f a = *(const v2f*)(&As[wave * 16 + lm][kk]);       // ds_load_b64
      const int kp = (ks << 1) + half;       // k pair for this lane half
#pragma unroll
      for (int tc = 0; tc < 8; ++tc) {
        v2f b = *(const v2f*)(&BsP[kp * 256 + (tc * 16 + lm) * 2]);  // ds_load_b64
        acc[tc] = __builtin_amdgcn_wmma_f32_16x16x4_f32(
            false, a, false, b, (short)0, acc[tc], false, false);
      }
    }
  }

  // Epilogue: C = alpha*acc + diag*I.  C layout: vgpr r -> row r + 8*half.
#pragma unroll
  for (int tc = 0; tc < 8; ++tc) {
#pragma unroll
    for (int vr = 0; vr < 8; ++vr) {
      int row = wave * 16 + vr + half * 8;
      int col = tc * 16 + lm;
      float v = alpha * acc[tc][vr];
      if (row == col) v += diag;
      Cg[(size_t)row * DN + col] = v;
    }
  }
}

// ---------------------------------------------------------------------------
// Elementwise / reduction helpers
// ---------------------------------------------------------------------------
__global__ void k_recond(const float* __restrict__ X, float* __restrict__ Xc, int total) {
  for (int i = blockIdx.x * blockDim.x + threadIdx.x; i < total; i += gridDim.x * blockDim.x) {
    int e = i % DD;
    float v = 0.999f * X[i];
    if ((e / DN) == (e % DN)) v += 0.001f;
    Xc[i] = v;
  }
}

// out[j] = mean over n of in[n*B*DD + j], j in [0, B*DD)
__global__ void k_mean(const float* __restrict__ in, float* __restrict__ out, int total) {
  for (int j = blockIdx.x * blockDim.x + threadIdx.x; j < total; j += gridDim.x * blockDim.x) {
    float s = 0.f;
    for (int n = 0; n < NN; ++n) s += in[(size_t)n * (BB * DD) + j];
    out[j] = s * (1.0f / NN);
  }
}

// Frobenius norm per matrix (one block per matrix)
__global__ void k_frob(const float* __restrict__ A, float* __restrict__ nrm) {
  __shared__ float red[256];
  size_t base = (size_t)blockIdx.x * DD;
  float s = 0.f;
  for (int i = threadIdx.x; i < DD; i += 256) {
    float v = A[base + i];
    s += v * v;
  }
  red[threadIdx.x] = s;
  __syncthreads();
  for (int w = 128; w > 0; w >>= 1) {
    if ((int)threadIdx.x < w) red[threadIdx.x] += red[threadIdx.x + w];
    __syncthreads();
  }
  if (threadIdx.x == 0) nrm[blockIdx.x] = sqrtf(red[0]);
}

// Newton-Schulz init: Y = A/s, Z = I
__global__ void k_ns_init(const float* __restrict__ A, const float* __restrict__ nrm,
                          float* __restrict__ Y, float* __restrict__ Z, int total) {
  for (int i = blockIdx.x * blockDim.x + threadIdx.x; i < total; i += gridDim.x * blockDim.x) {
    int m = i / DD, e = i % DD;
    float inv = 1.0f / nrm[m];
    Y[i] = A[i] * inv;
    Z[i] = ((e / DN) == (e % DN)) ? 1.f : 0.f;
  }
}

// Newton-Schulz finish: sqrt(A) = sqrt(s)*Y, invsqrt(A) = Z/sqrt(s)
__global__ void k_ns_fin(const float* __restrict__ Y, const float* __restrict__ Z,
                         const float* __restrict__ nrm, float* __restrict__ sq,
                         float* __restrict__ isq, int total) {
  for (int i = blockIdx.x * blockDim.x + threadIdx.x; i < total; i += gridDim.x * blockDim.x) {
    int m = i / DD;
    float rs = sqrtf(nrm[m]);
    if (sq)  sq[i]  = Y[i] * rs;
    if (isq) isq[i] = Z[i] / rs;
  }
}

__global__ void k_subI(const float* __restrict__ A, float* __restrict__ E, int total) {
  for (int i = blockIdx.x * blockDim.x + threadIdx.x; i < total; i += gridDim.x * blockDim.x) {
    int e = i % DD;
    E[i] = A[i] - (((e / DN) == (e % DN)) ? 1.f : 0.f);
  }
}

__global__ void k_setI(float* __restrict__ Q, float val, int total) {
  for (int i = blockIdx.x * blockDim.x + threadIdx.x; i < total; i += gridDim.x * blockDim.x) {
    int e = i % DD;
    Q[i] = ((e / DN) == (e % DN)) ? val : 0.f;
  }
}

__global__ void k_scale(const float* __restrict__ A, float a, float* __restrict__ O, int total) {
  for (int i = blockIdx.x * blockDim.x + threadIdx.x; i < total; i += gridDim.x * blockDim.x)
    O[i] = a * A[i];
}

__global__ void k_copy(const float* __restrict__ A, float* __restrict__ O, int total) {
  for (int i = blockIdx.x * blockDim.x + threadIdx.x; i < total; i += gridDim.x * blockDim.x)
    O[i] = A[i];
}

// ---------------------------------------------------------------------------
// Host orchestration (deterministic; stream-only; graph-capture safe)
// ---------------------------------------------------------------------------
extern "C" void kernel_launch(void* const* d_in, const int* in_sizes, int n_in,
                              void* d_out, int out_size, void* d_ws, size_t ws_size,
                              hipStream_t stream) {
  (void)in_sizes; (void)n_in; (void)out_size; (void)ws_size;
  const float* X = (const float*)d_in[0];
  float* Out = (float*)d_out;

  char* wsb = (char*)d_ws;
  size_t off = 0;
  auto alloc = [&](size_t nfloats) -> float* {
    float* p = (float*)(wsb + off);
    off += nfloats * sizeof(float);
    off = (off + 255) & ~(size_t)255;
    return p;
  };
  const size_t BIG = (size_t)NBM * DD;   // 2048 matrices
  const size_t SML = (size_t)BB * DD;    // 32 matrices

  float* Xc   = alloc(BIG);
  float* P0   = alloc(BIG);
  float* P1   = alloc(BIG);
  float* P2   = alloc(BIG);
  float* P3   = alloc(BIG);
  float* nrmN = alloc(NBM);
  float* nrmB = alloc(BB);
  float* refA = alloc(SML); float* refB = alloc(SML);
  float* sqB  = alloc(SML); float* isqB = alloc(SML);
  float* Yb   = alloc(SML); float* Zb   = alloc(SML);
  float* Wb   = alloc(SML); float* Ub   = alloc(SML);
  float* Tb   = alloc(SML);

  const int totB = (int)BIG, totS = (int)SML;
  auto g = [](int total) { return dim3((unsigned)((total + 255) / 256)); };
  auto bg = [&](const float* A, const float* Bm, float* Cm, int nmat, int modA,
                int modB, float alpha, float diag) {
    k_bgemm<<<dim3((unsigned)nmat), dim3(256), 0, stream>>>(A, Bm, Cm, modA, modB, alpha, diag);
  };

  // Xc = 0.999*X + 0.001*I ; ref = mean_n Xc
  k_recond<<<g(totB), 256, 0, stream>>>(X, Xc, totB);
  k_mean<<<g(totS), 256, 0, stream>>>(Xc, refA, totS);

  float* refCur = refA;
  float* refNxt = refB;

  for (int it = 0; it < 10; ++it) {
    // ---- sqrt/invsqrt of ref via Newton-Schulz (32 matrices) ----
    k_frob<<<dim3(BB), 256, 0, stream>>>(refCur, nrmB);
    k_ns_init<<<g(totS), 256, 0, stream>>>(refCur, nrmB, Yb, Zb, totS);
    {
      float *y = Yb, *z = Zb, *w = Wb, *u = Ub;
      for (int ns = 0; ns < 16; ++ns) {
        bg(z, y, w, BB, 0, 0, -0.5f, 1.5f);  // T = 1.5I - 0.5 ZY
        bg(y, w, u, BB, 0, 0, 1.f, 0.f);     // Y' = Y T
        bg(w, z, y, BB, 0, 0, 1.f, 0.f);     // Z' = T Z  (into freed buf)
        float* t0 = y; y = u; u = z; z = t0;
      }
      k_ns_fin<<<g(totS), 256, 0, stream>>>(y, z, nrmB, sqB, isqB, totS);
    }

    // ---- whiten: S = isqrt @ Xc @ isqrt  (2048 matrices) ----
    bg(isqB, Xc, P1, NBM, BB, 0, 1.f, 0.f);
    bg(P1, isqB, P0, NBM, 0, BB, 1.f, 0.f);

    // ---- logm(S): 4 NS square roots, then 16*log(I+E) by Horner ----
    float* rcur = P0; float* f1 = P1; float* f2 = P2; float* f3 = P3;
    for (int d2 = 0; d2 < 4; ++d2) {
      k_frob<<<dim3(NBM), 256, 0, stream>>>(rcur, nrmN);
      k_ns_init<<<g(totB), 256, 0, stream>>>(rcur, nrmN, f1, f2, totB);
      float *y = f1, *z = f2, *w = f3, *u = rcur;  // S buffer freed after init
      for (int ns = 0; ns < 12; ++ns) {
        bg(z, y, w, NBM, 0, 0, -0.5f, 1.5f);
        bg(y, w, u, NBM, 0, 0, 1.f, 0.f);
        bg(w, z, y, NBM, 0, 0, 1.f, 0.f);
        float* t0 = y; y = u; u = z; z = t0;
      }
      k_ns_fin<<<g(totB), 256, 0, stream>>>(y, z, nrmN, u, (float*)nullptr, totB);
      rcur = u; f1 = y; f2 = z; f3 = w;
    }
    k_subI<<<g(totB), 256, 0, stream>>>(rcur, f1, totB);         // E = R - I
    k_setI<<<g(totB), 256, 0, stream>>>(f2, -1.0f / 8.0f, totB); // innermost c8
    float* q = f2; float* qn = f3;
    for (int j = 7; j >= 1; --j) {
      float cj = ((j & 1) ? 1.f : -1.f) / (float)j;
      bg(f1, q, qn, NBM, 0, 0, 1.f, cj);                         // Q = E Q + c_j I
      float* t0 = q; q = qn; qn = t0;
    }
    bg(f1, q, rcur, NBM, 0, 0, 16.f, 0.f);                       // logS = 2^4 * E Q

    // ---- T = mean_n logS ----
    k_mean<<<g(totS), 256, 0, stream>>>(rcur, Tb, totS);

    // ---- expm(T): scale by 1/32, 8-term Taylor (Horner), square 5x ----
    k_scale<<<g(totS), 256, 0, stream>>>(Tb, 1.0f / 32.0f, Yb, totS);
    k_setI<<<g(totS), 256, 0, stream>>>(Zb, 1.0f, totS);
    float* eq = Zb; float* eqn = Wb;
    for (int j = 8; j >= 1; --j) {
      bg(Yb, eq, eqn, BB, 0, 0, 1.0f / (float)j, 1.0f);          // Q = (X/j) Q + I
      float* t0 = eq; eq = eqn; eqn = t0;
    }
    for (int sq = 0; sq < 5; ++sq) {
      bg(eq, eq, eqn, BB, 0, 0, 1.f, 0.f);
      float* t0 = eq; eq = eqn; eqn = t0;
    }

    // ---- new ref = sqrt @ expT @ sqrt ----
    bg(sqB, eq, Ub, BB, 0, 0, 1.f, 0.f);
    bg(Ub, sqB, refNxt, BB, 0, 0, 1.f, 0.f);
    float* t0 = refCur; refCur = refNxt; refNxt = t0;
  }

  k_copy<<<g(totS), 256, 0, stream>>>(refCur, Out, totS);
}